// Fusion_71700184039581
// MI455X (gfx1250) — compile-verified
//
#include <hip/hip_runtime.h>

#define KN    1024
#define EN    30000
#define EDIR  65536
#define EUND  65536
#define EEK   120000

typedef float v2f __attribute__((ext_vector_type(2)));
typedef float v8f __attribute__((ext_vector_type(8)));

// ---------- reduction helpers (256 threads = 8 wave32 waves) ----------
__device__ __forceinline__ float block_reduce_sum(float v, float* sbuf) {
    int lane = threadIdx.x & 31;
    int w    = threadIdx.x >> 5;
    #pragma unroll
    for (int o = 16; o; o >>= 1) v += __shfl_down(v, o);
    if (lane == 0) sbuf[w] = v;
    __syncthreads();
    if (threadIdx.x == 0) {
        float s = 0.f;
        #pragma unroll
        for (int i = 0; i < 8; ++i) s += sbuf[i];
        sbuf[0] = s;
    }
    __syncthreads();
    return sbuf[0];
}

__device__ __forceinline__ float block_reduce_max(float v, float* sbuf) {
    int lane = threadIdx.x & 31;
    int w    = threadIdx.x >> 5;
    #pragma unroll
    for (int o = 16; o; o >>= 1) v = fmaxf(v, __shfl_down(v, o));
    if (lane == 0) sbuf[w] = v;
    __syncthreads();
    if (threadIdx.x == 0) {
        float s = sbuf[0];
        #pragma unroll
        for (int i = 1; i < 8; ++i) s = fmaxf(s, sbuf[i]);
        sbuf[0] = s;
    }
    __syncthreads();
    return sbuf[0];
}

// ---------- WMMA fp32 GEMM: Z = A(1024x1024) @ B(1024x1024) ----------
// One wave per 16x16 output tile; K-loop in steps of 4 via V_WMMA_F32_16X16X4_F32.
__global__ __launch_bounds__(256) void gemm1024_f32(const float* __restrict__ A,
                                                    const float* __restrict__ B,
                                                    float* __restrict__ Z) {
    int wave = threadIdx.x >> 5;              // 0..7
    int lane = threadIdx.x & 31;
    int gw   = blockIdx.x * 8 + wave;         // 0..4095 (64x64 tiles)
    int tm   = (gw & 63) << 4;                // row base
    int tn   = (gw >> 6) << 4;                // col base
    int l16  = lane & 15;
    int kh   = (lane >> 4) << 1;              // 0 or 2 (K pair within step)

    v8f acc = {};
    const float* arow = A + (size_t)(tm + l16) * KN;
    for (int k0 = 0; k0 < KN; k0 += 4) {
        v2f a = *(const v2f*)(arow + k0 + kh);
        v2f b;
        b.x = B[(size_t)(k0 + kh) * KN + tn + l16];
        b.y = B[(size_t)(k0 + kh + 1) * KN + tn + l16];
        acc = __builtin_amdgcn_wmma_f32_16x16x4_f32(false, a, false, b,
                                                    (short)0, acc, false, false);
    }
    int row = tm + ((lane >> 4) << 3);
    int col = tn + l16;
    #pragma unroll
    for (int r = 0; r < 8; ++r) Z[(size_t)(row + r) * KN + col] = acc[r];
}

// ---------- row-wise dot products ----------
__global__ __launch_bounds__(256) void rowdot1(const float* __restrict__ M,
                                               const float* __restrict__ v,
                                               float* __restrict__ out, int rows) {
    __shared__ float sb[8];
    int r = blockIdx.x;
    if (r >= rows) return;
    const float* row = M + (size_t)r * KN;
    float p = 0.f;
    for (int t = threadIdx.x; t < KN; t += 256) p += row[t] * v[t];
    float s = block_reduce_sum(p, sb);
    if (threadIdx.x == 0) out[r] = s;
}

__global__ __launch_bounds__(256) void rowdot2(const float* __restrict__ M,
                                               const float* __restrict__ v1,
                                               const float* __restrict__ v2,
                                               float* __restrict__ o1,
                                               float* __restrict__ o2, int rows) {
    __shared__ float sb1[8], sb2[8];
    int r = blockIdx.x;
    if (r >= rows) return;
    const float* row = M + (size_t)r * KN;
    float p1 = 0.f, p2 = 0.f;
    for (int t = threadIdx.x; t < KN; t += 256) {
        float x = row[t];
        p1 += x * v1[t];
        p2 += x * v2[t];
    }
    float s1 = block_reduce_sum(p1, sb1);
    float s2 = block_reduce_sum(p2, sb2);
    if (threadIdx.x == 0) { o1[r] = s1; o2[r] = s2; }
}

// ---------- CSR build (group edges by destination) ----------
__global__ __launch_bounds__(256) void hist_dst(const int* __restrict__ dst,
                                                int* __restrict__ counts, int ne) {
    int e = blockIdx.x * 256 + threadIdx.x;
    if (e < ne) atomicAdd(counts + dst[e], 1);
}

// Single-block chunked scan: offsets[i+1] = inclusive prefix, offsets[0] = 0.
// Also rewrites counts[i] := exclusive prefix (becomes the scatter cursor).
__global__ __launch_bounds__(256) void scan_csr(int* __restrict__ counts,
                                                int* __restrict__ offsets, int n) {
    __shared__ int tmp[256];
    int run = 0;
    for (int base = 0; base < n; base += 256) {
        int i = base + threadIdx.x;
        int v = (i < n) ? counts[i] : 0;
        tmp[threadIdx.x] = v;
        __syncthreads();
        #pragma unroll
        for (int o = 1; o < 256; o <<= 1) {
            int t = (threadIdx.x >= o) ? tmp[threadIdx.x - o] : 0;
            __syncthreads();
            tmp[threadIdx.x] += t;
            __syncthreads();
        }
        int incl = tmp[threadIdx.x];
        int tot  = tmp[255];
        if (i < n) {
            offsets[i + 1] = run + incl;
            counts[i]      = run + incl - v;   // cursor = exclusive prefix
        }
        run += tot;
        __syncthreads();
    }
    if (threadIdx.x == 0) offsets[0] = 0;
}

__global__ __launch_bounds__(256) void scatter_csr(const int* __restrict__ dst,
                                                   int* __restrict__ cursor,
                                                   int* __restrict__ perm, int ne) {
    int e = blockIdx.x * 256 + threadIdx.x;
    if (e >= ne) return;
    int p = atomicAdd(cursor + dst[e], 1);
    perm[p] = e;
}

// ---------- fused per-destination GAT: softmax over in-edges + aggregation ----------
// Block = one dst node. No atomics: segment max/sum via block reductions,
// output row accumulated in registers and written once (zero for empty segments).
__global__ __launch_bounds__(256) void gat_aggregate(const int* __restrict__ src,
                                                     const int* __restrict__ offsets,
                                                     const int* __restrict__ perm,
                                                     const float* __restrict__ ssrc,
                                                     const float* __restrict__ sdst,
                                                     const float* __restrict__ z,
                                                     int srcoff,
                                                     float* __restrict__ out) {
    __shared__ float sbm[8], sbs[8];
    int d   = blockIdx.x;
    int beg = offsets[d], end = offsets[d + 1];
    float sd = sdst[d];

    // segment max
    float lm = -INFINITY;
    for (int idx = beg + threadIdx.x; idx < end; idx += 256) {
        int e = perm[idx];
        lm = fmaxf(lm, ssrc[src[e] - srcoff] + sd);
    }
    float m = block_reduce_max(lm, sbm);

    // segment sum of exp
    float ls = 0.f;
    for (int idx = beg + threadIdx.x; idx < end; idx += 256) {
        int e = perm[idx];
        ls += __expf(ssrc[src[e] - srcoff] + sd - m);
    }
    float den    = block_reduce_sum(ls, sbs);
    float invden = (end > beg) ? 1.f / den : 0.f;

    // weighted aggregation: 4 dims per thread, serial over the edge bucket
    float a0 = 0.f, a1 = 0.f, a2 = 0.f, a3 = 0.f;
    int t = threadIdx.x;
    for (int idx = beg; idx < end; ++idx) {
        int e = perm[idx];
        int s = src[e] - srcoff;
        float alpha = __expf(ssrc[s] + sd - m) * invden;
        const float* zr = z + (size_t)s * KN;
        a0 += alpha * zr[t];
        a1 += alpha * zr[t + 256];
        a2 += alpha * zr[t + 512];
        a3 += alpha * zr[t + 768];
    }
    float* orow = out + (size_t)d * KN;
    orow[t]       = a0;
    orow[t + 256] = a1;
    orow[t + 512] = a2;
    orow[t + 768] = a3;
}

// ---------- final fusions ----------
__global__ __launch_bounds__(256) void kn_fuse(const float* __restrict__ A,
                                               const float* __restrict__ Bm,
                                               const float* __restrict__ Cm,
                                               const float* __restrict__ kw1,
                                               const float* __restrict__ kb1,
                                               const float* __restrict__ kw2,
                                               const float* __restrict__ kb2,
                                               float* __restrict__ out) {
    __shared__ float sb1[8], sb2[8];
    __shared__ float sc0s, sc1s;
    int i = blockIdx.x;                       // 0..1023
    const float* ar = A + (size_t)i * KN;
    const float* br = Bm + (size_t)i * KN;
    const float* cr = Cm + (size_t)i * KN;
    float p1 = 0.f, p2 = 0.f;
    for (int t = threadIdx.x; t < KN; t += 256) {
        float a = ar[t], b = br[t], c = cr[t];
        p1 += a * kw1[t] + b * kw1[KN + t];
        p2 += a * kw2[t] + c * kw2[KN + t];
    }
    float S1 = block_reduce_sum(p1, sb1);
    float S2 = block_reduce_sum(p2, sb2);
    if (threadIdx.x == 0) {
        float s1 = S1 + kb1[0], s2 = S2 + kb2[0];
        float mm = fmaxf(s1, s2);
        float e1 = __expf(s1 - mm), e2 = __expf(s2 - mm);
        float inv = 1.f / (e1 + e2);
        sc0s = e1 * inv;
        sc1s = e2 * inv;
    }
    __syncthreads();
    float* orow = out + (size_t)i * KN;
    for (int t = threadIdx.x; t < KN; t += 256)
        orow[t] = ar[t] + sc0s * br[t] + sc1s * cr[t];
}

// Be lives in the exer region of d_out; finalize in place.
__global__ __launch_bounds__(256) void exer_fuse(const float* __restrict__ X,
                                                 const float* __restrict__ ew1,
                                                 const float* __restrict__ eb1,
                                                 float* __restrict__ out) {
    __shared__ float sb[8];
    __shared__ float se_s;
    int j = blockIdx.x;                       // 0..EN-1
    const float* xr = X + (size_t)j * KN;
    float* brow = out + (size_t)j * KN;
    float p = 0.f;
    for (int t = threadIdx.x; t < KN; t += 256)
        p += xr[t] * ew1[t] + brow[t] * ew1[KN + t];
    float S = block_reduce_sum(p, sb);
    if (threadIdx.x == 0) se_s = S + eb1[0];
    __syncthreads();
    for (int t = threadIdx.x; t < KN; t += 256) {
        float b = brow[t];
        brow[t] = xr[t] + se_s * b;
    }
}

// ---------- host launcher ----------
extern "C" void kernel_launch(void* const* d_in, const int* in_sizes, int n_in,
                              void* d_out, int out_size, void* d_ws, size_t ws_size,
                              hipStream_t stream) {
    const float* exer   = (const float*)d_in[0];
    const float* kn     = (const float*)d_in[1];
    const int*   dsrc   = (const int*)d_in[2];
    const int*   ddst   = (const int*)d_in[3];
    const int*   usrc   = (const int*)d_in[4];
    const int*   udst   = (const int*)d_in[5];
    const int*   esrc   = (const int*)d_in[6];
    const int*   edst   = (const int*)d_in[7];
    const float* W_dir  = (const float*)d_in[8];
    const float* a_dir  = (const float*)d_in[9];
    const float* W_und  = (const float*)d_in[10];
    const float* a_und  = (const float*)d_in[11];
    const float* W_ek   = (const float*)d_in[12];
    const float* a_ek   = (const float*)d_in[13];
    const float* k_w1   = (const float*)d_in[14];
    const float* k_b1   = (const float*)d_in[15];
    const float* k_w2   = (const float*)d_in[16];
    const float* k_b2   = (const float*)d_in[17];
    const float* e_w1   = (const float*)d_in[18];
    const float* e_b1   = (const float*)d_in[19];

    float* out = (float*)d_out;
    float* ws  = (float*)d_ws;

    const size_t M1 = (size_t)KN * KN;        // 1048576
    size_t off = 0;
    float* zdir  = ws + off; off += M1;
    float* zund  = ws + off; off += M1;
    float* zek   = ws + off; off += M1;
    float* Bacc  = ws + off; off += M1;
    float* Cacc  = ws + off; off += M1;
    float* wv    = ws + off; off += KN;
    float* sdi_s = ws + off; off += KN;
    float* sdi_d = ws + off; off += KN;
    float* sun_s = ws + off; off += KN;
    float* sun_d = ws + off; off += KN;
    float* sek_s = ws + off; off += KN;
    float* sek_d = ws + off; off += EN;
    // CSR buffers: the three count arrays are contiguous -> one memset.
    size_t zbeg = off;
    int* cnt_d = (int*)(ws + off); off += KN;
    int* cnt_u = (int*)(ws + off); off += KN;
    int* cnt_e = (int*)(ws + off); off += EN;
    size_t zend = off;
    int* off_d = (int*)(ws + off); off += KN + 1;
    int* off_u = (int*)(ws + off); off += KN + 1;
    int* off_e = (int*)(ws + off); off += EN + 1;
    int* perm_d = (int*)(ws + off); off += EDIR;
    int* perm_u = (int*)(ws + off); off += EUND;
    int* perm_e = (int*)(ws + off); off += EEK;

    hipMemsetAsync(ws + zbeg, 0, (zend - zbeg) * sizeof(float), stream);

    // Three 1024^3 fp32 WMMA GEMMs (only knowledge-node z-rows are ever needed).
    gemm1024_f32<<<512, 256, 0, stream>>>(kn, W_dir, zdir);
    gemm1024_f32<<<512, 256, 0, stream>>>(kn, W_und, zund);
    gemm1024_f32<<<512, 256, 0, stream>>>(kn, W_ek,  zek);

    // Attention score vectors.
    rowdot1<<<KN, 256, 0, stream>>>(W_ek, a_ek + KN, wv, KN);   // wv = W_ek @ a_ek[K:]
    rowdot2<<<KN, 256, 0, stream>>>(zdir, a_dir, a_dir + KN, sdi_s, sdi_d, KN);
    rowdot2<<<KN, 256, 0, stream>>>(zund, a_und, a_und + KN, sun_s, sun_d, KN);
    rowdot1<<<KN, 256, 0, stream>>>(zek, a_ek, sek_s, KN);      // src-side (kn nodes)
    rowdot1<<<EN, 256, 0, stream>>>(exer, wv, sek_d, EN);       // dst-side (exer nodes)

    // CSR build for all three graphs.
    hist_dst<<<(EDIR + 255) / 256, 256, 0, stream>>>(ddst, cnt_d, EDIR);
    hist_dst<<<(EUND + 255) / 256, 256, 0, stream>>>(udst, cnt_u, EUND);
    hist_dst<<<(EEK  + 255) / 256, 256, 0, stream>>>(edst, cnt_e, EEK);
    scan_csr<<<1, 256, 0, stream>>>(cnt_d, off_d, KN);
    scan_csr<<<1, 256, 0, stream>>>(cnt_u, off_u, KN);
    scan_csr<<<1, 256, 0, stream>>>(cnt_e, off_e, EN);
    scatter_csr<<<(EDIR + 255) / 256, 256, 0, stream>>>(ddst, cnt_d, perm_d, EDIR);
    scatter_csr<<<(EUND + 255) / 256, 256, 0, stream>>>(udst, cnt_u, perm_u, EUND);
    scatter_csr<<<(EEK  + 255) / 256, 256, 0, stream>>>(edst, cnt_e, perm_e, EEK);

    // Fused softmax + aggregation, atomic-free, one block per destination row.
    gat_aggregate<<<KN, 256, 0, stream>>>(dsrc, off_d, perm_d, sdi_s, sdi_d, zdir, 0,  Bacc);
    gat_aggregate<<<KN, 256, 0, stream>>>(usrc, off_u, perm_u, sun_s, sun_d, zund, 0,  Cacc);
    gat_aggregate<<<EN, 256, 0, stream>>>(esrc, off_e, perm_e, sek_s, sek_d, zek,  EN, out);

    // Final fusions.
    kn_fuse<<<KN, 256, 0, stream>>>(kn, Bacc, Cacc, k_w1, k_b1, k_w2, k_b2,
                                    out + (size_t)EN * KN);
    exer_fuse<<<EN, 256, 0, stream>>>(exer, e_w1, e_b1, out);
}